// PartBasedDecoomposition_30382598652046
// MI455X (gfx1250) — compile-verified
//
#include <hip/hip_runtime.h>

#define BS 64
#define NPTS 16384
#define NSEG 16
#define NFPS 256
#define CAP_PER_BATCH 16640   // 16384 + 16*16 padding slack, per batch
#define NTHREADS 1024

typedef float v2f __attribute__((ext_vector_type(2)));
typedef float v8f __attribute__((ext_vector_type(8)));

// ---------------------------------------------------------------------------
// Kernel 1: per-batch stable partition of points by segment label.
// Compacted layout per batch: segments contiguous, each padded to multiple of
// 16 (zero-filled pads) so every 16-row WMMA tile is segment-homogeneous.
// Stored as float4 {x, y, z, |p|^2}. Also writes counts/offsets and masks.
// ---------------------------------------------------------------------------
__global__ __launch_bounds__(NTHREADS)
void fps_compact_kernel(const float* __restrict__ x,
                        float4* __restrict__ comp,
                        int* __restrict__ g_counts,
                        int* __restrict__ g_offsets,
                        float* __restrict__ out_masks)
{
    __shared__ int cnt[NSEG];
    __shared__ int offs[NSEG];
    __shared__ int base[NSEG];
    __shared__ int waveCnt[32 * NSEG];

    const int b    = blockIdx.x;
    const int tid  = threadIdx.x;
    const int lane = tid & 31;
    const int wave = tid >> 5;

    const float4* xb4 = (const float4*)(x + (size_t)b * NPTS * 4);
    float4* cb = comp + (size_t)b * CAP_PER_BATCH;

    if (tid < NSEG) cnt[tid] = 0;
    __syncthreads();

    // ---- count per segment via wave32 ballots ----
    int acc = 0;  // lane s (<16) accumulates this wave's count for segment s
    for (int c = 0; c < NPTS / NTHREADS; ++c) {
        float4 pv = xb4[c * NTHREADS + tid];
        int lab = (int)pv.w;
        lab = (lab < 0) ? 0 : (lab > 15 ? 15 : lab);
        #pragma unroll
        for (int s = 0; s < NSEG; ++s) {
            unsigned m = (unsigned)__ballot(lab == s);
            if (lane == s) acc += __popc(m);
        }
    }
    if (lane < NSEG) atomicAdd(&cnt[lane], acc);
    __syncthreads();

    if (tid == 0) {
        int run = 0;
        for (int s = 0; s < NSEG; ++s) {
            offs[s] = run;
            run += (cnt[s] + 15) & ~15;          // padded to tile multiple
            g_counts[b * NSEG + s]  = cnt[s];
            g_offsets[b * NSEG + s] = offs[s];
            out_masks[b * NSEG + s] = (cnt[s] >= NFPS) ? 1.0f : 0.0f;
        }
    }
    if (tid < NSEG) base[tid] = 0;
    __syncthreads();

    // ---- stable scatter (preserves original index order per segment) ----
    for (int c = 0; c < NPTS / NTHREADS; ++c) {
        float4 pv = xb4[c * NTHREADS + tid];
        int lab = (int)pv.w;
        lab = (lab < 0) ? 0 : (lab > 15 ? 15 : lab);
        unsigned myBal = 0;
        #pragma unroll
        for (int s = 0; s < NSEG; ++s) {
            unsigned m = (unsigned)__ballot(lab == s);
            if (lab == s) myBal = m;
            if (lane == s) waveCnt[wave * NSEG + s] = __popc(m);
        }
        int rankInWave = __popc(myBal & ((1u << lane) - 1u));
        __syncthreads();
        int prefix = 0;
        for (int w = 0; w < wave; ++w) prefix += waveCnt[w * NSEG + lab];
        int dst = offs[lab] + base[lab] + prefix + rankInWave;
        float4 v;
        v.x = pv.x; v.y = pv.y; v.z = pv.z;
        v.w = pv.x * pv.x + pv.y * pv.y + pv.z * pv.z;   // |p|^2
        cb[dst] = v;
        __syncthreads();
        if (tid < NSEG) {
            int tot = 0;
            for (int w = 0; w < 32; ++w) tot += waveCnt[w * NSEG + tid];
            base[tid] += tot;
        }
        __syncthreads();
    }

    // ---- zero-fill padding slots ----
    for (int s = 0; s < NSEG; ++s) {
        int pc = (((cnt[s] + 15) & ~15)) - cnt[s];
        for (int i = tid; i < pc; i += NTHREADS) {
            float4 z0; z0.x = 0.f; z0.y = 0.f; z0.z = 0.f; z0.w = 0.f;
            cb[offs[s] + cnt[s] + i] = z0;
        }
    }
}

// ---------------------------------------------------------------------------
// Kernel 2: one workgroup per batch runs 256 sequential FPS steps for all 16
// segments simultaneously. Squared distances via V_WMMA_F32_16X16X4_F32:
//   D[s][j] = (-2 q_s) . p_j + |p_j|^2 + |q_s|^2 = |p_j - q_s|^2
// A (16x4): row s = [-2qx, -2qy, -2qz, 1];  B (4x16): col j = [x, y, z, |p|^2]
// C: |q_s|^2 broadcast per row. min_d kept in LDS (66.5 KB — CDNA5 big-LDS).
// Per-tile metadata packs {segment, #valid lanes} so the hot loop needs no
// extra LDS reads; segment id goes through readfirstlane -> M0-indexed
// v_movrels extraction + one ds_bpermute.
// ---------------------------------------------------------------------------
__global__ __launch_bounds__(NTHREADS)
void fps_iter_kernel(const float4* __restrict__ comp,
                     const int* __restrict__ g_counts,
                     const int* __restrict__ g_offsets,
                     float* __restrict__ out)
{
    __shared__ float minD[CAP_PER_BATCH];
    __shared__ int tileInfo[CAP_PER_BATCH / 16];   // s | (validLanes << 8)
    __shared__ float4 qs[NSEG];
    __shared__ unsigned long long slots[NSEG];
    __shared__ int cnts[NSEG], offs[NSEG];
    __shared__ int nTilesSh;

    const int b    = blockIdx.x;
    const int tid  = threadIdx.x;
    const int lane = tid & 31;
    const int wave = tid >> 5;
    const int ll   = lane & 15;

    const float4* cb = comp + (size_t)b * CAP_PER_BATCH;
    float* gp = out + (size_t)(b * NSEG) * NFPS * 3;

    if (tid < NSEG) {
        cnts[tid]  = g_counts[b * NSEG + tid];
        offs[tid]  = g_offsets[b * NSEG + tid];
        slots[tid] = 0ull;
    }
    __syncthreads();
    if (tid == 0)
        nTilesSh = (offs[NSEG - 1] + ((cnts[NSEG - 1] + 15) & ~15)) >> 4;
    if (tid < NSEG) {
        float4 q; q.x = 0.f; q.y = 0.f; q.z = 0.f; q.w = 0.f;
        if (cnts[tid] > 0) q = cb[offs[tid]];   // FPS starts at first valid point
        qs[tid] = q;
    }
    __syncthreads();
    const int ntiles = nTilesSh;

    for (int i = tid; i < ntiles * 16; i += NTHREADS) minD[i] = __builtin_inff();
    for (int t = tid; t < ntiles; t += NTHREADS) {
        int p = t * 16, s = 0;
        #pragma unroll
        for (int k = 1; k < NSEG; ++k) if (p >= offs[k]) s = k;
        int valid = offs[s] + cnts[s] - p;               // lanes with real points
        valid = (valid < 0) ? 0 : (valid > 16 ? 16 : valid);
        tileInfo[t] = s | (valid << 8);
    }
    __syncthreads();

    for (int it = 0; it < NFPS; ++it) {
        // ---- emit current selections (point BEFORE argmax update) ----
        if (tid < NSEG * 3) {
            int s = tid / 3, c = tid % 3;
            float v = 0.0f;
            if (cnts[s] >= NFPS) {
                float4 q = qs[s];
                v = (c == 0) ? q.x : (c == 1) ? q.y : q.z;
            }
            gp[(s * NFPS + it) * 3 + c] = v;
        }
        __syncthreads();

        // ---- build A (queries) and C (|q|^2) operands, wave-wide ----
        float4 qa = qs[ll];
        float A0 = (lane < 16) ? -2.0f * qa.x : -2.0f * qa.z;
        float A1 = (lane < 16) ? -2.0f * qa.y : 1.0f;
        v2f Am; Am[0] = A0; Am[1] = A1;
        v8f Cm;
        int hb = (lane < 16) ? 0 : 8;
        #pragma unroll
        for (int v = 0; v < 8; ++v) Cm[v] = qs[v + hb].w;

        // ---- distance tiles: one WMMA per 16 points vs all 16 queries ----
        auto doTile = [&](int t) {
            // packed {seg, valid} is wave-uniform; readfirstlane => scalar ops
            int info  = __builtin_amdgcn_readfirstlane(tileInfo[t]);
            int s     = info & 0xFF;
            int valid = info >> 8;
            float4 pt = cb[t * 16 + ll];
            v2f Bm;
            Bm[0] = (lane < 16) ? pt.x : pt.y;
            Bm[1] = (lane < 16) ? pt.z : pt.w;
            v8f D = __builtin_amdgcn_wmma_f32_16x16x4_f32(
                false, Am, false, Bm, (short)0, Cm, false, false);
            float dv;
            switch (s & 7) {
                case 0:  dv = D[0]; break;
                case 1:  dv = D[1]; break;
                case 2:  dv = D[2]; break;
                case 3:  dv = D[3]; break;
                case 4:  dv = D[4]; break;
                case 5:  dv = D[5]; break;
                case 6:  dv = D[6]; break;
                default: dv = D[7]; break;
            }
            float dist = __shfl(dv, ll | ((s >> 3) << 4), 32);
            if (lane < valid) {                 // scalar-bound guard, no LDS reads
                int gi = t * 16 + lane;
                minD[gi] = fminf(minD[gi], dist);
            }
        };

        // 2x unrolled: two independent WMMAs in flight per trip
        for (int t = wave; t < ntiles; t += 64) {
            if (t + 64 < ntiles)
                __builtin_prefetch((const void*)(cb + (t + 64) * 16 + ll), 0, 1);
            doTile(t);
            int t2 = t + 32;
            if (t2 < ntiles) doTile(t2);
        }
        __syncthreads();

        // ---- per-segment masked argmax (2 waves per segment) ----
        {
            int s   = wave >> 1;
            int h   = wave & 1;
            int cnt = cnts[s], off = offs[s];
            const float2* m2p = (const float2*)(minD + off);   // off is 16-aligned
            unsigned long long best = 0ull;
            for (int j = h * 32 + lane; 2 * j < cnt; j += 64) {
                float2 m2 = m2p[j];                            // ds_load_b64
                int li0 = 2 * j;
                unsigned long long k0 =
                    ((unsigned long long)__float_as_uint(m2.x) << 32) |
                    (unsigned)(0x7FFFFFFF - li0);
                best = (best > k0) ? best : k0;
                if (li0 + 1 < cnt) {
                    unsigned long long k1 =
                        ((unsigned long long)__float_as_uint(m2.y) << 32) |
                        (unsigned)(0x7FFFFFFF - (li0 + 1));
                    best = (best > k1) ? best : k1;
                }
            }
            if (best) atomicMax(&slots[s], best);   // ds_max_u64 in LDS
        }
        __syncthreads();

        // ---- decode winners into next queries, reset slots ----
        if (tid < NSEG) {
            int cnt = cnts[tid];
            if (cnt > 0) {
                unsigned long long k = slots[tid];
                int li = 0x7FFFFFFF - (int)(unsigned)(k & 0xFFFFFFFFu);
                if (li >= 0 && li < cnt) qs[tid] = cb[offs[tid] + li];
                slots[tid] = 0ull;
            }
        }
        __syncthreads();
    }
}

// ---------------------------------------------------------------------------
extern "C" void kernel_launch(void* const* d_in, const int* in_sizes, int n_in,
                              void* d_out, int out_size, void* d_ws, size_t ws_size,
                              hipStream_t stream)
{
    (void)in_sizes; (void)n_in; (void)out_size; (void)ws_size;
    const float* x = (const float*)d_in[0];
    float* out = (float*)d_out;

    char* ws = (char*)d_ws;
    int*    g_counts  = (int*)(ws);
    int*    g_offsets = (int*)(ws + 4096);
    float4* comp      = (float4*)(ws + 65536);   // 64 * 16640 * 16B ≈ 17 MB

    float* masks = out + (size_t)BS * NSEG * NFPS * 3;

    hipLaunchKernelGGL(fps_compact_kernel, dim3(BS), dim3(NTHREADS), 0, stream,
                       x, comp, g_counts, g_offsets, masks);
    hipLaunchKernelGGL(fps_iter_kernel, dim3(BS), dim3(NTHREADS), 0, stream,
                       comp, g_counts, g_offsets, out);
}